// MultiHeadAttentionQuantum_65481071401275
// MI455X (gfx1250) — compile-verified
//
#include <hip/hip_runtime.h>
#include <math.h>

typedef float v2f __attribute__((ext_vector_type(2)));
typedef float v4f __attribute__((ext_vector_type(4)));
typedef float v8f __attribute__((ext_vector_type(8)));

#define SEQ   4096
#define EMB   8
#define BATCH 8
#define KTILES (SEQ / 16)       // 256
#define LDS_STRIDE 20           // padded row stride (even -> b64-aligned float2 loads)

// One wave (32 lanes) handles one 16-row query tile. Scores are bounded
// (|s| <= 8/sqrt(8) ~ 2.83) so softmax needs no max subtraction; row sums are
// produced by a ones-column in V through the same P@V WMMAs. The 1/sqrt(d)
// scale is folded into the Q operand. All matrix math in exact f32 via
// V_WMMA_F32_16X16X4_F32.
__global__ __launch_bounds__(256) void qattn_fused_kernel(
    const float* __restrict__ x,      // [B, S, 8]
    const float* __restrict__ theta,  // [8]
    const float* __restrict__ W,      // [8, 8]  (out, in)
    float* __restrict__ out)          // [B, S, 8]
{
    __shared__ float lds_p[8 * 16 * LDS_STRIDE];   // per-wave P-transpose slab

    const int lane  = threadIdx.x & 31;
    const int wave  = threadIdx.x >> 5;
    const int gtile = blockIdx.x * 8 + wave;       // 0 .. 2047
    const int b     = gtile >> 8;                  // tile / (S/16)
    const int qt    = gtile & (KTILES - 1);
    const int r     = lane & 15;                   // row (A) / col (B,C,D) index
    const int half  = lane >> 4;

    const float* __restrict__ xb = x + (size_t)b * (SEQ * EMB);
    float* pl = lds_p + wave * (16 * LDS_STRIDE);

    // theta values this lane needs for A/B chunk loads (cols 4j + 2*half + {0,1})
    float thc[2][2];
#pragma unroll
    for (int j = 0; j < 2; ++j) {
        thc[j][0] = theta[4 * j + 2 * half];
        thc[j][1] = theta[4 * j + 2 * half + 1];
    }
    const int   nc   = r & 7;                  // clamped d-column for V operand
    const float thn  = theta[nc];
    const float vsel = (r < 8)  ? 1.0f : 0.0f; // real V columns 0..7
    const float ones = (r == 8) ? 1.0f : 0.0f; // rowsum column (n = 8)

    const float scale = 0.35355339059327373f;  // 1/sqrt(8), folded into Q

    // ---- Q tile in A-layout: lane (r,half) holds {q[r][4j+2h], q[r][4j+2h+1]} ----
    const int qrow0 = qt * 16;
    v2f aq[2];
#pragma unroll
    for (int j = 0; j < 2; ++j) {
        v2f xv = *(const v2f*)(xb + (qrow0 + r) * EMB + 4 * j + 2 * half);
        aq[j][0] = scale * __cosf(xv[0] + thc[j][0]);
        aq[j][1] = scale * __cosf(xv[1] + thc[j][1]);
    }

    v8f acc = {};   // 16x16 C/D layout: cols 0..7 = P@V, col 8 = rowsum(P), rest 0

    for (int kt = 0; kt < KTILES; ++kt) {
        const int t0 = kt * 16;

        // K tile as B operand: B = K^T, whose layout mirrors A from row-major K
        v2f bk[2];
#pragma unroll
        for (int j = 0; j < 2; ++j) {
            v2f xv = *(const v2f*)(xb + (t0 + r) * EMB + 4 * j + 2 * half);
            bk[j][0] = __cosf(xv[0] + thc[j][0]);
            bk[j][1] = __cosf(xv[1] + thc[j][1]);
        }

        // S = (Q/sqrt(d)) K^T  (d=8 -> two K=4 WMMAs, chained through C)
        v8f c = {};
        c = __builtin_amdgcn_wmma_f32_16x16x4_f32(false, aq[0], false, bk[0],
                                                  (short)0, c, false, false);
        c = __builtin_amdgcn_wmma_f32_16x16x4_f32(false, aq[1], false, bk[1],
                                                  (short)0, c, false, false);

        // P = exp(S); bounded scores -> no max shift needed
#pragma unroll
        for (int i = 0; i < 8; ++i)
            c[i] = __expf(c[i]);

        // ---- transpose P (C/D layout -> A layout) through per-wave LDS slab ----
#pragma unroll
        for (int i = 0; i < 8; ++i)
            pl[(i + 8 * half) * LDS_STRIDE + r] = c[i];
        asm volatile("s_wait_dscnt 0" ::: "memory");

        // acc += P @ [V | 1 | 0...]   (K=16 -> four K=4 WMMAs)
#pragma unroll
        for (int j = 0; j < 4; ++j) {
            v2f ap = *(const v2f*)(pl + r * LDS_STRIDE + 4 * j + 2 * half);
            v2f bv;
            bv[0] = vsel * __cosf(xb[(t0 + 4 * j + 2 * half    ) * EMB + nc] + thn) + ones;
            bv[1] = vsel * __cosf(xb[(t0 + 4 * j + 2 * half + 1) * EMB + nc] + thn) + ones;
            acc = __builtin_amdgcn_wmma_f32_16x16x4_f32(false, ap, false, bv,
                                                        (short)0, acc, false, false);
        }
    }

    // ---- stage attention numerator + denominator (col 8) through LDS ----
#pragma unroll
    for (int i = 0; i < 8; ++i)
        pl[(i + 8 * half) * LDS_STRIDE + r] = acc[i];
    asm volatile("s_wait_dscnt 0" ::: "memory");

    // ---- output projection: out[row] = (num[row]/den[row]) @ W^T, 2 lanes/row ----
    const int orow = lane >> 1;
    const int eb   = (lane & 1) * 4;
    const float inv = 1.0f / pl[orow * LDS_STRIDE + 8];
    float o0 = 0.f, o1 = 0.f, o2 = 0.f, o3 = 0.f;
#pragma unroll
    for (int n = 0; n < 8; ++n) {
        float a = pl[orow * LDS_STRIDE + n];
        o0 += a * W[(eb + 0) * 8 + n];
        o1 += a * W[(eb + 1) * 8 + n];
        o2 += a * W[(eb + 2) * 8 + n];
        o3 += a * W[(eb + 3) * 8 + n];
    }
    v4f ov = {o0 * inv, o1 * inv, o2 * inv, o3 * inv};
    *(v4f*)(out + ((size_t)b * SEQ + qrow0 + orow) * EMB + eb) = ov;
}

extern "C" void kernel_launch(void* const* d_in, const int* in_sizes, int n_in,
                              void* d_out, int out_size, void* d_ws, size_t ws_size,
                              hipStream_t stream) {
    (void)in_sizes; (void)n_in; (void)out_size; (void)d_ws; (void)ws_size;
    const float* x     = (const float*)d_in[0];  // [8, 4096, 8]
    const float* theta = (const float*)d_in[1];  // [8]
    const float* W_out = (const float*)d_in[2];  // [8, 8]
    float* out = (float*)d_out;                  // [8, 4096, 8]

    // 2048 query tiles of 16 rows; 8 waves per block -> 256 blocks of 256 threads
    dim3 grid(BATCH * KTILES / 8);
    dim3 block(256);
    hipLaunchKernelGGL(qattn_fused_kernel, grid, block, 0, stream,
                       x, theta, W_out, out);
}